// GHMRLoss_16183436771679
// MI455X (gfx1250) — compile-verified
//
#include <hip/hip_runtime.h>
#include <hip/hip_bf16.h>
#include <math.h>

// GHMR loss:  result = (1/N) * sum_b w_b * S_b
//   S_b     = sum of per-sample loss with clipped bin b
//   count_b = # samples with raw bin == b (valid: g in [0,1))
//   w_b     = max(count_b,1)^(-0.75)
//
// Memory floor: 134 MB @ 23.3 TB/s ~= 5.8 us. To stay under it on the compute
// side we use the CDNA5 hardware transcendental V_TANH_F32 (one trans op per
// value instead of ~15-20 VALU ops of libm tanh), and the CDNA5 async copy
// path global_load_async_to_lds_b128 (ASYNCcnt) with a double-buffered LDS
// pipeline + s_wait_asynccnt.

#define GHMR_MU    0.02f
#define GHMR_BINS  10

static constexpr int T      = 256;          // threads per block (8 wave32)
static constexpr int TILE   = T * 4;        // 1024 floats per tile (16B/thread)
static constexpr int TILES  = 8;            // tiles per block
static constexpr int EPB    = TILE * TILES; // 8192 elements per block
static constexpr int HSTR   = 22;           // hist stride (floats): 10 float2 bins + 2 pad
                                            // 22 mod 64 banks -> conflict-free over 32 lanes

// ---- CDNA5 async global->LDS copy (16 bytes), GV addressing mode ----------
__device__ __forceinline__ void async_copy16(unsigned lds_byte_off, const void* gptr) {
    unsigned long long ga = (unsigned long long)(uintptr_t)gptr;
    asm volatile("global_load_async_to_lds_b128 %0, %1, off"
                 :
                 : "v"(lds_byte_off), "v"(ga)
                 : "memory");
}
__device__ __forceinline__ void wait_async_le2() {
    asm volatile("s_wait_asynccnt 2" ::: "memory");
}
__device__ __forceinline__ void wait_async_0() {
    asm volatile("s_wait_asynccnt 0" ::: "memory");
}

// ---- CDNA5 hardware tanh (trans op). ISA hazard rule: 1 independent op or
// V_NOP after a trans op before its output is used / sources overwritten.
// In the 8-wide block each tanh covers the previous one; trailing v_nops
// cover the last.
__device__ __forceinline__ void tanh8(const float4& a, const float4& b,
                                      float* ta, float* tb) {
    asm volatile(
        "v_tanh_f32 %0, %8\n\t"
        "v_tanh_f32 %1, %9\n\t"
        "v_tanh_f32 %2, %10\n\t"
        "v_tanh_f32 %3, %11\n\t"
        "v_tanh_f32 %4, %12\n\t"
        "v_tanh_f32 %5, %13\n\t"
        "v_tanh_f32 %6, %14\n\t"
        "v_tanh_f32 %7, %15\n\t"
        "v_nop\n\t"
        "v_nop"
        : "=&v"(ta[0]), "=&v"(ta[1]), "=&v"(ta[2]), "=&v"(ta[3]),
          "=&v"(tb[0]), "=&v"(tb[1]), "=&v"(tb[2]), "=&v"(tb[3])
        : "v"(a.x), "v"(a.y), "v"(a.z), "v"(a.w),
          "v"(b.x), "v"(b.y), "v"(b.z), "v"(b.w));
}
__device__ __forceinline__ float tanh_hw(float x) {
    float r;
    asm volatile("v_tanh_f32 %0, %1\n\tv_nop\n\tv_nop" : "=&v"(r) : "v"(x));
    return r;
}

// One element: update this thread's private interleaved {loss_sum, count} bins.
__device__ __forceinline__ void ghmr_update(float p, float t, float tp, float tt,
                                            float2* hrow) {
    float d    = fabsf(p - t);
    float loss = (d < GHMR_MU) ? (d * d * (0.5f / GHMR_MU)) : (d - 0.5f * GHMR_MU);
    float g    = fabsf(tp - tt);
    int   raw  = (int)(g * (float)GHMR_BINS);            // g >= 0: trunc == floor
    int   bin  = raw < (GHMR_BINS - 1) ? raw : (GHMR_BINS - 1);
    float vf   = (raw < GHMR_BINS) ? 1.0f : 0.0f;        // valid iff g in [0,1)
    float2 v   = hrow[bin];                              // ds_load_b64
    v.x += loss;
    v.y += vf;
    hrow[bin] = v;                                       // ds_store_b64
}

__global__ __launch_bounds__(T)
void ghmr_partials_kernel(const float* __restrict__ pred,
                          const float* __restrict__ targ,
                          float* __restrict__ ws) {
    __shared__ __align__(16) float s_pred[2][TILE];      // 8 KB
    __shared__ __align__(16) float s_targ[2][TILE];      // 8 KB
    __shared__ __align__(16) float s_hist[T * HSTR];     // 22 KB, {sum,cnt} pairs

    const int tid  = threadIdx.x;
    const int base = blockIdx.x * EPB;

    float2* hrow = (float2*)&s_hist[tid * HSTR];
#pragma unroll
    for (int b = 0; b < GHMR_BINS; ++b) hrow[b] = make_float2(0.0f, 0.0f);

    const unsigned lp0 = (unsigned)(uintptr_t)&s_pred[0][tid * 4];
    const unsigned lp1 = (unsigned)(uintptr_t)&s_pred[1][tid * 4];
    const unsigned lt0 = (unsigned)(uintptr_t)&s_targ[0][tid * 4];
    const unsigned lt1 = (unsigned)(uintptr_t)&s_targ[1][tid * 4];

    // Prologue: tile 0 -> stage 0 (ASYNCcnt = 2 outstanding).
    async_copy16(lp0, pred + base + tid * 4);
    async_copy16(lt0, targ + base + tid * 4);

    for (int t = 0; t < TILES; ++t) {
        if (t + 1 < TILES) {
            const int nb = base + (t + 1) * TILE + tid * 4;
            if (((t + 1) & 1) == 0) { async_copy16(lp0, pred + nb); async_copy16(lt0, targ + nb); }
            else                    { async_copy16(lp1, pred + nb); async_copy16(lt1, targ + nb); }
            wait_async_le2();        // tile t's two loads complete (in-order)
        } else {
            wait_async_0();
        }
        const int s = t & 1;
        float4 p4 = *(const float4*)&s_pred[s][tid * 4];
        float4 t4 = *(const float4*)&s_targ[s][tid * 4];

        float tp[4], tt[4];
        tanh8(p4, t4, tp, tt);       // 8x v_tanh_f32 (trans pipe)

        ghmr_update(p4.x, t4.x, tp[0], tt[0], hrow);
        ghmr_update(p4.y, t4.y, tp[1], tt[1], hrow);
        ghmr_update(p4.z, t4.z, tp[2], tt[2], hrow);
        ghmr_update(p4.w, t4.w, tp[3], tt[3], hrow);
    }

    __syncthreads();

    // Deterministic block reduction: thread b < 10 owns bin b.
    if (tid < GHMR_BINS) {
        float s = 0.0f, c = 0.0f;
        for (int i = 0; i < T; ++i) {
            s += s_hist[i * HSTR + tid * 2];
            c += s_hist[i * HSTR + tid * 2 + 1];
        }
        ws[blockIdx.x * (2 * GHMR_BINS) + tid]             = s;
        ws[blockIdx.x * (2 * GHMR_BINS) + GHMR_BINS + tid] = c;
    }
}

// Single-block finalize: reduce per-block partials in fixed order (deterministic),
// fold in tail elements, apply w_b = tot^(-0.75), write mean.
__global__ __launch_bounds__(32)
void ghmr_finalize_kernel(const float* __restrict__ ws, int nblocks,
                          const float* __restrict__ pred,
                          const float* __restrict__ targ,
                          int tail_start, int n,
                          float* __restrict__ out) {
    __shared__ __align__(16) float s_hist[32 * HSTR];
    __shared__ float s_part[GHMR_BINS];

    const int tid = threadIdx.x;
    float2* hrow = (float2*)&s_hist[tid * HSTR];
#pragma unroll
    for (int b = 0; b < GHMR_BINS; ++b) hrow[b] = make_float2(0.0f, 0.0f);

    // Tail elements not covered by full blocks (none when N % 8192 == 0).
    for (int i = tail_start + tid; i < n; i += 32) {
        float p = pred[i], t = targ[i];
        ghmr_update(p, t, tanh_hw(p), tanh_hw(t), hrow);
    }

    __syncthreads();

    if (tid < GHMR_BINS) {
        double s = 0.0, c = 0.0;
        for (int i = 0; i < nblocks; ++i) {
            s += (double)ws[i * (2 * GHMR_BINS) + tid];
            c += (double)ws[i * (2 * GHMR_BINS) + GHMR_BINS + tid];
        }
        for (int i = 0; i < 32; ++i) {
            s += (double)s_hist[i * HSTR + tid * 2];
            c += (double)s_hist[i * HSTR + tid * 2 + 1];
        }
        float tot = fmaxf((float)c, 1.0f);
        float w   = exp2f(-0.75f * log2f(tot));          // tot^(-0.75)
        s_part[tid] = w * (float)s;
    }
    __syncthreads();

    if (tid == 0) {
        double acc = 0.0;
        for (int b = 0; b < GHMR_BINS; ++b) acc += (double)s_part[b];
        out[0] = (float)(acc / (double)n);
    }
}

extern "C" void kernel_launch(void* const* d_in, const int* in_sizes, int n_in,
                              void* d_out, int out_size, void* d_ws, size_t ws_size,
                              hipStream_t stream) {
    const float* pred = (const float*)d_in[0];
    const float* targ = (const float*)d_in[1];
    float* out = (float*)d_out;
    float* ws  = (float*)d_ws;

    const int n       = in_sizes[0];   // 16777216 for the reference
    const int nblocks = n / EPB;       // 2048 blocks (160 KB of ws)
    const int tail    = nblocks * EPB; // == n when N % 8192 == 0

    if (nblocks > 0)
        ghmr_partials_kernel<<<nblocks, T, 0, stream>>>(pred, targ, ws);
    ghmr_finalize_kernel<<<1, 32, 0, stream>>>(ws, nblocks, pred, targ, tail, n, out);
}